// SuperSloMo_Engine_46239617909217
// MI455X (gfx1250) — compile-verified
//
#include <hip/hip_runtime.h>

// Backward bilinear warp (SuperSloMo): out[b,c,y,x] = bilinear(img[b,c], y+fy, x+fx)
// Memory-bound: ~236 MB min traffic -> ~10 us floor at 23.3 TB/s HBM.
// img (88.5 MB) fits in the 192 MB L2: img gathers use default RT hint and stay
// resident; flow/out are single-touch streams: non-temporal B128 so they don't
// evict img. 3D grid removes all integer-divide index math (uniform b/y in SGPRs).

typedef float v4f __attribute__((ext_vector_type(4)));

#define BB 8
#define CC 3
#define HH 720
#define WW 1280

__global__ __launch_bounds__(320) void backwarp_b128_nt(
    const float* __restrict__ img,
    const float* __restrict__ flow,
    float* __restrict__ out)
{
    constexpr int HW = HH * WW;

    // blockDim.x == 320 == WW/4 quads: one block = one image row. No bounds check,
    // no divides; b*2*HW + y*WW is wave-uniform -> SGPR base, lane offset in VGPR.
    const int xq = threadIdx.x << 2;     // leftmost pixel of this 4-pixel quad
    const int y  = blockIdx.y;
    const int b  = blockIdx.z;

    // ---- streaming flow loads: global_load_b128 th:NT (single use, keep out of L2) ----
    const float* __restrict__ flowRow = flow + (size_t)b * 2 * HW + (size_t)y * WW;
    const v4f fx = __builtin_nontemporal_load((const v4f*)(flowRow + xq));
    const v4f fy = __builtin_nontemporal_load((const v4f*)(flowRow + HW + xq));

    const float* __restrict__ imgB = img + (size_t)b * (CC * HW);

    v4f o0, o1, o2;

    #pragma unroll
    for (int i = 0; i < 4; ++i) {
        // Reference's normalize->denormalize is an exact algebraic identity:
        // sample coord = clamp(pixel + flow, 0, dim-1).
        float xf = (float)(xq + i) + fx[i];
        float yf = (float)y        + fy[i];
        xf = fminf(fmaxf(xf, 0.0f), (float)(WW - 1));
        yf = fminf(fmaxf(yf, 0.0f), (float)(HH - 1));

        const float x0f = floorf(xf);
        const float y0f = floorf(yf);
        const float wx  = xf - x0f;
        const float wy  = yf - y0f;

        const int x0 = (int)x0f;
        const int y0 = (int)y0f;
        const int x1 = min(x0 + 1, WW - 1);
        const int y1 = min(y0 + 1, HH - 1);

        const int i00 = y0 * WW + x0;
        const int i01 = y0 * WW + x1;
        const int i10 = y1 * WW + x0;
        const int i11 = y1 * WW + x1;

        // 4 taps x 3 planar channels; spatially local (flow ~ N(0,1)) -> L2 hits,
        // img is fully L2-resident after first touch.
        #pragma unroll
        for (int c = 0; c < CC; ++c) {
            const float* __restrict__ p = imgB + c * HW;
            const float v00 = p[i00];
            const float v01 = p[i01];
            const float v10 = p[i10];
            const float v11 = p[i11];
            const float top = v00 + wx * (v01 - v00);
            const float bot = v10 + wx * (v11 - v10);
            const float r   = top + wy * (bot - top);
            if      (c == 0) o0[i] = r;
            else if (c == 1) o1[i] = r;
            else             o2[i] = r;
        }
    }

    // ---- streaming output stores: global_store_b128 th:NT ----
    float* __restrict__ outRow = out + (size_t)b * (CC * HW) + (size_t)y * WW;
    __builtin_nontemporal_store(o0, (v4f*)(outRow + xq));
    __builtin_nontemporal_store(o1, (v4f*)(outRow + HW + xq));
    __builtin_nontemporal_store(o2, (v4f*)(outRow + 2 * HW + xq));
}

extern "C" void kernel_launch(void* const* d_in, const int* in_sizes, int n_in,
                              void* d_out, int out_size, void* d_ws, size_t ws_size,
                              hipStream_t stream) {
    (void)in_sizes; (void)n_in; (void)out_size; (void)d_ws; (void)ws_size;
    const float* img  = (const float*)d_in[0];
    const float* flow = (const float*)d_in[1];
    float*       out  = (float*)d_out;

    const dim3 block(WW / 4, 1, 1);      // 320 threads = 10 wave32 waves = 1 row
    const dim3 grid(1, HH, BB);          // 5760 blocks
    hipLaunchKernelGGL(backwarp_b128_nt, grid, block, 0, stream, img, flow, out);
}